// LinearCausalAttention_64939905516282
// MI455X (gfx1250) — compile-verified
//
#include <hip/hip_runtime.h>

typedef __attribute__((ext_vector_type(2))) float v2f;
typedef __attribute__((ext_vector_type(8))) float v8f;

namespace {
constexpr int B_     = 2;
constexpr int NH_    = 8;
constexpr int T_     = 2048;
constexpr int N_     = 256;
constexpr int D_     = 256;
constexpr int NTT    = T_ / 16;   // 128 tiles along T
constexpr int HALF_N = N_ / 2;    // 128 rope frequency pairs

constexpr int VSTR2 = 272;  // V pair-row stride (float2): 2*272 % 64 == 32 -> half-waves on disjoint bank halves
constexpr int KSTR2 = 130;  // Kr row stride (float2): word stride 260 % 64 == 4 -> lane*4 bank stripe
constexpr int SSTR  = 18;   // S-stage row stride (float): 18*m % 64 distinct for m=0..15
}

// ---------------------------------------------------------------------------
// Kernel 0: rope cos/sin table, tab[t][i] = (cos(t*f_i), sin(t*f_i)).
// 2048*128 float2 = 2 MB at d_ws+0.
// ---------------------------------------------------------------------------
__global__ void rope_table_kernel(float2* __restrict__ tab) {
    int idx = blockIdx.x * blockDim.x + threadIdx.x;
    if (idx >= T_ * HALF_N) return;
    int t = idx / HALF_N;
    int i = idx - t * HALF_N;
    float ex   = -((float)(2 * i) / (float)N_) * 13.28771237954945f; // log2(10000)
    float freq = exp2f(ex);
    float ang  = (float)t * freq;
    float s, c;
    sincosf(ang, &s, &c);
    tab[idx] = make_float2(c, s);
}

// ---------------------------------------------------------------------------
// Kernel 0b (big-ws path): pre-rotate Q and K into workspace.
// Flat pair id = ((bh*T + t)*128 + pi).
// ---------------------------------------------------------------------------
__global__ void rope_apply_kernel(const float*  __restrict__ Q,
                                  const float*  __restrict__ K,
                                  const float2* __restrict__ tab,
                                  float* __restrict__ Qr,
                                  float* __restrict__ Kr) {
    size_t idx = (size_t)blockIdx.x * blockDim.x + threadIdx.x;
    if (idx >= (size_t)B_ * NH_ * T_ * HALF_N) return;
    int pi = (int)(idx & (HALF_N - 1));
    int t  = (int)((idx >> 7) & (T_ - 1));
    float2 cs = tab[(size_t)t * HALF_N + pi];
    size_t off = idx * 2;
    float2 q = *(const float2*)(Q + off);
    float2 k = *(const float2*)(K + off);
    *(float2*)(Qr + off) = make_float2(q.x * cs.x - q.y * cs.y,
                                       q.x * cs.y + q.y * cs.x);
    *(float2*)(Kr + off) = make_float2(k.x * cs.x - k.y * cs.y,
                                       k.x * cs.y + k.y * cs.x);
}

// Load an aligned rope pair (k, k+1) of a row from global and rotate it.
__device__ __forceinline__ v2f rope_load(const float* __restrict__ row,
                                         const float2* __restrict__ tabRow,
                                         int k) {
    float2 x  = *(const float2*)(row + k);
    float2 cs = tabRow[k >> 1];
    v2f r;
    r.x = x.x * cs.x - x.y * cs.y;
    r.y = x.x * cs.y + x.y * cs.x;
    return r;
}

__device__ __forceinline__ v2f load2(const float* __restrict__ p) {
    float2 x = *(const float2*)p;
    v2f r; r.x = x.x; r.y = x.y;
    return r;
}

__device__ __forceinline__ v8f wmma4(v2f a, v2f b, v8f c) {
    return __builtin_amdgcn_wmma_f32_16x16x4_f32(
        false, a, false, b, (short)0, c, false, false);
}

// ---------------------------------------------------------------------------
// Main kernel: block = 8 waves, one (b,h).  Wave w owns t-tile (tg + 16*w);
// waves march over s-tiles in lockstep; V (pair-interleaved) and K tiles are
// staged cooperatively in LDS once per step.
// PREROPED: Q/K args already rotated (pure copies), else rotate in-loop.
// ---------------------------------------------------------------------------
template <bool PREROPED>
__global__ void __launch_bounds__(256)
lca_main_kernel(const float*  __restrict__ Q,
                const float*  __restrict__ K,
                const float*  __restrict__ V,
                const float2* __restrict__ tab,
                float* __restrict__ out,
                float* __restrict__ scores) {
    __shared__ float2 vt2[8 * VSTR2];     // V tile, rows pair-interleaved  17.4 KB
    __shared__ float2 kt2[16 * KSTR2];    // Kr tile                        16.6 KB
    __shared__ float  st[8 * 16 * SSTR];  // per-wave S-tile stage           9 KB

    const int lane = threadIdx.x & 31;
    const int wave = threadIdx.x >> 5;

    const int bh      = blockIdx.x >> 4;
    const int tg      = blockIdx.x & 15;
    const int myTile  = tg + wave * 16;
    const int maxTile = tg + 112;
    const int b       = bh >> 3;
    const int t0      = myTile * 16;

    const int col  = lane & 15;
    const int half = lane >> 4;
    const int koff = half * 2;

    const float* Qbh = Q + (size_t)bh * T_ * N_;
    const float* Kbh = K + (size_t)bh * T_ * N_;
    const float* Vb  = V + (size_t)b  * T_ * D_;
    float* Obh = out    + (size_t)bh * T_ * D_;
    float* Sbh = scores + (size_t)bh * T_ * T_;

    const int     tq   = t0 + col;
    const float*  qrow = Qbh + (size_t)tq * N_;
    const float2* qtab = tab + (size_t)tq * HALF_N;

    v8f oacc[16];
#pragma unroll
    for (int i = 0; i < 16; ++i) oacc[i] = v8f{0,0,0,0,0,0,0,0};

    float* myst = st + wave * (16 * SSTR);

    for (int sTile = 0; sTile <= maxTile; ++sTile) {
        const int s0 = sTile * 16;

        if (sTile < maxTile) {
            __builtin_prefetch(Vb  + (size_t)(s0 + 16) * D_ + threadIdx.x * 16, 0, 0);
            __builtin_prefetch(Kbh + (size_t)(s0 + 16) * N_ + threadIdx.x * 16, 0, 0);
        }

        __syncthreads();   // previous step done consuming vt2/kt2

        // ---- cooperative stage: V tile, rows 2p/2p+1 interleaved ----
        {
#pragma unroll
            for (int j = 0; j < 8; ++j) {
                int e = threadIdx.x + j * 256;
                int p = e >> 8;
                int d = e & 255;
                float x = Vb[(size_t)(s0 + 2 * p)     * D_ + d];
                float y = Vb[(size_t)(s0 + 2 * p + 1) * D_ + d];
                vt2[p * VSTR2 + d] = make_float2(x, y);
            }
        }
        // ---- cooperative stage: K tile ----
        if constexpr (PREROPED) {
            const float4* src = (const float4*)(Kbh + (size_t)s0 * N_); // contiguous 16KB
#pragma unroll
            for (int j = 0; j < 4; ++j) {
                int id  = threadIdx.x + j * 256;
                int row = id >> 6;
                int c4  = id & 63;
                *(float4*)((float*)&kt2[row * KSTR2] + c4 * 4) = src[id];
            }
        } else {
#pragma unroll
            for (int j = 0; j < 8; ++j) {
                int pid  = threadIdx.x + j * 256;
                int row  = pid >> 7;
                int pi   = pid & 127;
                int srow = s0 + row;
                float2 x  = *(const float2*)(Kbh + (size_t)srow * N_ + pi * 2);
                float2 cs = tab[(size_t)srow * HALF_N + pi];
                kt2[row * KSTR2 + pi] =
                    make_float2(x.x * cs.x - x.y * cs.y,
                                x.x * cs.y + x.y * cs.x);
            }
        }
        __syncthreads();

        if (sTile <= myTile) {
            // ---- S tile = Qr * Kr^T, two independent wmma chains ----
            v8f c0 = v8f{0,0,0,0,0,0,0,0};
            v8f c1 = v8f{0,0,0,0,0,0,0,0};
            const float2* krow2 = kt2 + col * KSTR2;
#pragma unroll 4
            for (int kk = 0; kk < 64; kk += 2) {
                const int ka = kk * 4 + koff;
                const int kb = ka + 4;
                v2f a0, a1;
                if constexpr (PREROPED) {
                    a0 = load2(qrow + ka);
                    a1 = load2(qrow + kb);
                } else {
                    a0 = rope_load(qrow, qtab, ka);
                    a1 = rope_load(qrow, qtab, kb);
                }
                float2 k0v = krow2[ka >> 1];
                float2 k1v = krow2[kb >> 1];
                v2f b0; b0.x = k0v.x; b0.y = k0v.y;
                v2f b1; b1.x = k1v.x; b1.y = k1v.y;
                c0 = wmma4(a0, b0, c0);
                c1 = wmma4(a1, b1, c1);
            }
            v8f c = c0 + c1;

            // strict-lower causal mask on diagonal tile (keep s < t)
            if (sTile == myTile) {
#pragma unroll
                for (int r = 0; r < 8; ++r)
                    if (col >= r + half * 8) c[r] = 0.0f;
            }

            // write scores (non-temporal write-once stream) + stage tile
#pragma unroll
            for (int r = 0; r < 8; ++r) {
                const int tr = r + half * 8;
                __builtin_nontemporal_store(c[r],
                    Sbh + (size_t)(t0 + tr) * T_ + s0 + col);
                myst[tr * SSTR + col] = c[r];
            }

            // S tile as A fragments
            v2f aS[4];
#pragma unroll
            for (int k0 = 0; k0 < 4; ++k0)
                aS[k0] = load2(myst + col * SSTR + k0 * 4 + koff);

            // ---- O(16x256) += S * V: one ds_load_b64 per B fragment ----
#pragma unroll
            for (int k0 = 0; k0 < 4; ++k0) {
                const float2* vp = vt2 + (k0 * 2 + half) * VSTR2 + col;
                v2f bv[16];
#pragma unroll
                for (int dt = 0; dt < 16; ++dt) {
                    float2 w = vp[dt * 16];
                    bv[dt].x = w.x;
                    bv[dt].y = w.y;
                }
#pragma unroll
                for (int dt = 0; dt < 16; ++dt)
                    oacc[dt] = wmma4(aS[k0], bv[dt], oacc[dt]);
            }
        }
    }

    // zero the never-computed upper-triangle score tiles (d_out is poisoned)
    for (int sTile = myTile + 1; sTile < NTT; ++sTile) {
        const int s0 = sTile * 16;
#pragma unroll
        for (int r = 0; r < 8; ++r)
            __builtin_nontemporal_store(0.0f,
                Sbh + (size_t)(t0 + r + half * 8) * T_ + s0 + col);
    }

    // write the out stripe (non-temporal, write-once)
#pragma unroll
    for (int dt = 0; dt < 16; ++dt)
#pragma unroll
        for (int r = 0; r < 8; ++r)
            __builtin_nontemporal_store(oacc[dt][r],
                Obh + (size_t)(t0 + r + half * 8) * D_ + dt * 16 + col);
}

extern "C" void kernel_launch(void* const* d_in, const int* in_sizes, int n_in,
                              void* d_out, int out_size, void* d_ws, size_t ws_size,
                              hipStream_t stream) {
    const float* Q = (const float*)d_in[0];   // (2, 8, 2048, 256)
    const float* K = (const float*)d_in[1];   // (2, 8, 2048, 256)
    const float* V = (const float*)d_in[2];   // (2, 1, 2048, 256)

    float* out    = (float*)d_out;                        // (2,8,2048,256)
    float* scores = out + (size_t)B_ * NH_ * T_ * D_;     // (2,8,2048,2048)

    const size_t tabBytes = (size_t)T_ * HALF_N * sizeof(float2);   // 2 MB
    const size_t qkElems  = (size_t)B_ * NH_ * T_ * N_;             // 8.4M floats each

    float2* tab = (float2*)d_ws;
    const int tabElems = T_ * HALF_N;
    rope_table_kernel<<<(tabElems + 255) / 256, 256, 0, stream>>>(tab);

    const size_t needBytes = tabBytes + 2 * qkElems * sizeof(float);  // ~69 MB
    if (ws_size >= needBytes) {
        // big-workspace path: pre-rotate Q/K, hot loop is pure loads + WMMA
        float* qr = (float*)((char*)d_ws + tabBytes);
        float* kr = qr + qkElems;
        const size_t pairs = (size_t)B_ * NH_ * T_ * HALF_N;          // 4.2M
        rope_apply_kernel<<<(unsigned)((pairs + 255) / 256), 256, 0, stream>>>(
            Q, K, tab, qr, kr);
        lca_main_kernel<true><<<B_ * NH_ * 16, 256, 0, stream>>>(
            qr, kr, V, tab, out, scores);
    } else {
        // small-workspace fallback: rotate in-loop via the 2 MB table
        lca_main_kernel<false><<<B_ * NH_ * 16, 256, 0, stream>>>(
            Q, K, V, tab, out, scores);
    }
}